// CrossLayer_40596030882151
// MI455X (gfx1250) — compile-verified
//
#include <hip/hip_runtime.h>
#include <stdint.h>

#define D_DIM 4096
#define LN_EPS 1e-5f

typedef float v4f __attribute__((ext_vector_type(4)));
typedef float v2f __attribute__((ext_vector_type(2)));
typedef float v8f __attribute__((ext_vector_type(8)));

__device__ __forceinline__ float wave_reduce_add(float v) {
    // wave32 butterfly reduction
    v += __shfl_xor(v, 16, 32);
    v += __shfl_xor(v, 8, 32);
    v += __shfl_xor(v, 4, 32);
    v += __shfl_xor(v, 2, 32);
    v += __shfl_xor(v, 1, 32);
    return v;
}

// Sum 8 wave-partials (in LDS) and broadcast to all lanes using
// v_wmma_f32_16x16x4_f32 with B = all-ones (layout independent).
// A 16x4 fp32 layout (ISA 7.12.2): lanes 0-15 hold K0 (VGPR0) / K1 (VGPR1),
// lanes 16-31 hold K2 (VGPR0) / K3 (VGPR1). With every row m identical,
// D[m][n] = P0+P1+P2+P3 (+C). Two chained WMMAs sum all 8 partials.
__device__ __forceinline__ float block_sum8_wmma(const float* p, bool hi) {
    v2f a0; a0.x = hi ? p[2] : p[0]; a0.y = hi ? p[3] : p[1];
    v2f a1; a1.x = hi ? p[6] : p[4]; a1.y = hi ? p[7] : p[5];
    v2f ones; ones.x = 1.0f; ones.y = 1.0f;
    v8f c = {};
    c = __builtin_amdgcn_wmma_f32_16x16x4_f32(false, a0, false, ones, (short)0, c, false, false);
    c = __builtin_amdgcn_wmma_f32_16x16x4_f32(false, a1, false, ones, (short)0, c, false, false);
    return c[0]; // every element of D is the full broadcast sum
}

__global__ __launch_bounds__(256)
void fused_cross_ln_kernel(const float* __restrict__ x,
                           const float* __restrict__ x0,
                           const float* __restrict__ w,
                           const float* __restrict__ bias,
                           const float* __restrict__ gamma,
                           const float* __restrict__ beta,
                           float* __restrict__ out) {
    __shared__ __attribute__((aligned(16))) float s_x[D_DIM]; // 16 KB row stage
    __shared__ float s_red[6][8];                             // 6 stats x 8 waves

    const int b    = blockIdx.x;
    const int t    = threadIdx.x;
    const int lane = t & 31;
    const int wave = t >> 5;

    const float* xrow  = x  + (size_t)b * D_DIM;
    const float* x0row = x0 + (size_t)b * D_DIM;

    // ---- Phase 0: async-copy x row into LDS (CDNA5 ASYNCcnt path) ----
    const uint32_t lds_base = (uint32_t)(uintptr_t)(&s_x[0]);
    #pragma unroll
    for (int c = 0; c < 4; ++c) {
        const int e = (t + c * 256) * 4;                 // float index
        const uint32_t lds_addr = lds_base + (uint32_t)e * 4u;
        const float* g = xrow + e;
        asm volatile("global_load_async_to_lds_b128 %0, %1, off"
                     :: "v"(lds_addr), "v"(g) : "memory");
    }

    // ---- Phase 1 (overlapped with async copy): dot(x0,w) + bias stats ----
    float p_dot = 0.f, p_sb = 0.f, p_sbb = 0.f;
    v4f bb[4];
    #pragma unroll
    for (int c = 0; c < 4; ++c) {
        const int e = (t + c * 256) * 4;
        v4f a  = __builtin_nontemporal_load((const v4f*)(x0row + e)); // read-once stream
        v4f wv = *(const v4f*)(w + e);
        v4f bv = *(const v4f*)(bias + e);
        bb[c] = bv;
        p_dot += a.x * wv.x + a.y * wv.y + a.z * wv.z + a.w * wv.w;
        p_sb  += bv.x + bv.y + bv.z + bv.w;
        p_sbb += bv.x * bv.x + bv.y * bv.y + bv.z * bv.z + bv.w * bv.w;
    }

    // ---- Wait for the async x tile, then make it visible to all waves ----
    asm volatile("s_wait_asynccnt 0" ::: "memory");
    __syncthreads();

    // ---- Phase 2: x stats from LDS, keep x in registers for the write pass ----
    float p_sx = 0.f, p_sxx = 0.f, p_sxb = 0.f;
    v4f xv[4];
    #pragma unroll
    for (int c = 0; c < 4; ++c) {
        const int e = (t + c * 256) * 4;
        v4f v = *(const v4f*)(s_x + e);
        xv[c] = v;
        p_sx  += v.x + v.y + v.z + v.w;
        p_sxx += v.x * v.x + v.y * v.y + v.z * v.z + v.w * v.w;
        v4f bv = bb[c];
        p_sxb += v.x * bv.x + v.y * bv.y + v.z * bv.z + v.w * bv.w;
    }

    // ---- Block reduction: wave32 shuffles, then WMMA broadcast-sum ----
    p_dot = wave_reduce_add(p_dot);
    p_sx  = wave_reduce_add(p_sx);
    p_sxx = wave_reduce_add(p_sxx);
    p_sxb = wave_reduce_add(p_sxb);
    p_sb  = wave_reduce_add(p_sb);
    p_sbb = wave_reduce_add(p_sbb);
    if (lane == 0) {
        s_red[0][wave] = p_dot;
        s_red[1][wave] = p_sx;
        s_red[2][wave] = p_sxx;
        s_red[3][wave] = p_sxb;
        s_red[4][wave] = p_sb;
        s_red[5][wave] = p_sbb;
    }
    __syncthreads();

    const bool hi = lane >= 16;
    float S[6];
    #pragma unroll
    for (int q = 0; q < 6; ++q) {
        S[q] = block_sum8_wmma(&s_red[q][0], hi);
    }

    const float scale = S[0];                    // x0 . w
    const float a1p   = 1.0f + scale;            // pre = x*(1+scale) + bias
    const float invD  = 1.0f / (float)D_DIM;
    const float mean  = (a1p * S[1] + S[4]) * invD;
    const float msq   = (a1p * a1p * S[2] + 2.0f * a1p * S[3] + S[5]) * invD;
    const float var   = msq - mean * mean;
    const float rstd  = rsqrtf(var + LN_EPS);

    // ---- Phase 3: normalize + affine, streamed NT store ----
    float* orow = out + (size_t)b * D_DIM;
    #pragma unroll
    for (int c = 0; c < 4; ++c) {
        const int e = (t + c * 256) * 4;
        v4f g  = *(const v4f*)(gamma + e);
        v4f bt = *(const v4f*)(beta + e);
        v4f o  = (xv[c] * a1p + bb[c] - mean) * rstd * g + bt;
        __builtin_nontemporal_store(o, (v4f*)(orow + e));
    }
}

extern "C" void kernel_launch(void* const* d_in, const int* in_sizes, int n_in,
                              void* d_out, int out_size, void* d_ws, size_t ws_size,
                              hipStream_t stream) {
    const float* x     = (const float*)d_in[0];
    const float* x0    = (const float*)d_in[1];
    const float* w     = (const float*)d_in[2];
    const float* bias  = (const float*)d_in[3];
    const float* gamma = (const float*)d_in[4];
    const float* beta  = (const float*)d_in[5];
    float* out = (float*)d_out;

    const int Brows = in_sizes[0] / D_DIM; // 8192
    fused_cross_ln_kernel<<<Brows, 256, 0, stream>>>(x, x0, w, bias, gamma, beta, out);
}